// Decoder_52664888983802
// MI455X (gfx1250) — compile-verified
//
#include <hip/hip_runtime.h>

typedef __attribute__((ext_vector_type(2))) float v2f;
typedef __attribute__((ext_vector_type(4))) float v4f;
typedef __attribute__((ext_vector_type(8))) float v8f;

constexpr int kE           = 512;      // E
constexpr int kW           = 16;       // W
constexpr int kTilesPerBC  = 1000;     // 16000 frames / 16 frames per tile
constexpr int kOutPerBC    = 128008;   // 8*(16000-1)+16
constexpr int kWavesPerBlk = 8;        // 256 threads, wave32
constexpr int kTilesPerWav = 4;
constexpr int kTotalTiles  = 16000;    // 16 (b,c) * 1000

__global__ __launch_bounds__(256) void decoder_ola_wmma(
    const float* __restrict__ mixture,   // [256000, 512] row-major
    const float* __restrict__ weight,    // [16, 512] row-major
    float* __restrict__ out)             // [16, 128008], pre-zeroed
{
  __shared__ float wlds[kW * kE];                         // 32 KB weights
  __shared__ float stage[kWavesPerBlk][16 * 17 + 8];      // padded D tiles

  const int tid = threadIdx.x;

  // Cooperative load of the whole weight matrix into LDS (float4).
  {
    const v4f* src = (const v4f*)weight;
    v4f*       dst = (v4f*)wlds;
    for (int i = tid; i < (kW * kE) / 4; i += 256) dst[i] = src[i];
  }
  __syncthreads();

  const int wave = tid >> 5;
  const int lane = tid & 31;
  const int m    = lane & 15;   // A row within tile / B column (n)
  const int h    = lane >> 4;   // half-wave selector (hw K group)

  float* st = &stage[wave][0];

  const int tile0 = (blockIdx.x * kWavesPerBlk + wave) * kTilesPerWav;

  for (int t = 0; t < kTilesPerWav; ++t) {
    const int tile = tile0 + t;

    // Per-lane pointers. K is processed in permuted groups of 8 so that
    // each lane's A fetch is one contiguous b128 (see analysis).
    const float* Aptr = mixture + (size_t)(tile * 16 + m) * kE + 4 * h;
    const float* Bptr = wlds + m * kE + 4 * h;   // n == lane&15

    v8f c = {0.f, 0.f, 0.f, 0.f, 0.f, 0.f, 0.f, 0.f};

    #pragma unroll 4
    for (int k = 0; k < kE; k += 8) {
      v4f a = __builtin_nontemporal_load((const v4f*)(Aptr + k)); // streamed once
      v4f b = *(const v4f*)(Bptr + k);
      v2f a0 = {a.x, a.y}, a1 = {a.z, a.w};
      v2f b0 = {b.x, b.y}, b1 = {b.z, b.w};
      c = __builtin_amdgcn_wmma_f32_16x16x4_f32(false, a0, false, b0,
                                                (short)0, c, false, false);
      c = __builtin_amdgcn_wmma_f32_16x16x4_f32(false, a1, false, b1,
                                                (short)0, c, false, false);
    }

    // Stage D tile to LDS: lane holds D[r + 8h][m] in c[r] (ISA C/D layout).
    #pragma unroll
    for (int r = 0; r < 8; ++r)
      st[(r + 8 * h) * 17 + m] = c[r];

    // Overlap-add: tile covers output positions [tileBase, tileBase+136).
    const int bc = tile / kTilesPerBC;
    const int lt = tile % kTilesPerBC;
    float* obase = out + (size_t)bc * kOutPerBC + (size_t)lt * 128;

    #pragma unroll
    for (int i = 0; i < 4; ++i) {
      int p  = lane + 32 * i;          // 0..127
      int m0 = p >> 3, r = p & 7;
      float v = st[m0 * 17 + r];                      // frame m0, n = r
      if (m0 >= 1) v += st[(m0 - 1) * 17 + r + 8];    // frame m0-1, n = r+8
      if (p >= 8) obase[p] = v;                       // interior: exclusive
      else        atomicAdd(&obase[p], v);            // seam with prev tile
    }
    if (lane < 8) {                                   // seam with next tile
      atomicAdd(&obase[128 + lane], st[15 * 17 + lane + 8]);
    }
  }
}

extern "C" void kernel_launch(void* const* d_in, const int* in_sizes, int n_in,
                              void* d_out, int out_size, void* d_ws, size_t ws_size,
                              hipStream_t stream) {
  const float* mixture = (const float*)d_in[0];  // [8,2,16000,512] fp32
  const float* weight  = (const float*)d_in[1];  // [16,512] fp32
  float* out = (float*)d_out;                    // [8,2,128008] fp32

  // Seam positions are accumulated with atomicAdd -> need zeroed output.
  hipMemsetAsync(d_out, 0, (size_t)out_size * sizeof(float), stream);

  dim3 grid(kTotalTiles / (kWavesPerBlk * kTilesPerWav));  // 500 blocks
  decoder_ola_wmma<<<grid, 256, 0, stream>>>(mixture, weight, out);
}